// GenerateMask_2_0_13460427505896
// MI455X (gfx1250) — compile-verified
//
#include <hip/hip_runtime.h>
#include <math.h>

typedef __attribute__((ext_vector_type(2))) float v2f;
typedef __attribute__((ext_vector_type(8))) float v8f;

#define CDIM 256   // channels (reduction dim)
#define PIX  64    // h*w pixels
#define CS   260   // padded LDS stride for xt[p][c]  (260 % 64 == 4 -> conflict-free b64 B loads)

// ---------------------------------------------------------------------------
// Prep: wn2[k][c] = w2[k][c] / ||w2[k]||   (folds w_norm into the A operand)
// ---------------------------------------------------------------------------
__global__ void wnorm_prep_kernel(const float* __restrict__ w2,
                                  float* __restrict__ wn2) {
    const int k = blockIdx.x;
    const int t = threadIdx.x;          // 256 threads
    float v = w2[(size_t)k * CDIM + t];
    float s = v * v;
    #pragma unroll
    for (int off = 16; off; off >>= 1) s += __shfl_xor(s, off, 32);
    __shared__ float ps[8];
    __shared__ float inv;
    if ((t & 31) == 0) ps[t >> 5] = s;
    __syncthreads();
    if (t == 0) {
        float tot = 0.f;
        #pragma unroll
        for (int i = 0; i < 8; ++i) tot += ps[i];
        inv = 1.0f / sqrtf(tot);
    }
    __syncthreads();
    wn2[(size_t)k * CDIM + t] = v * inv;
}

// monotone float -> uint mapping (preserves total order; bijective)
__device__ __forceinline__ unsigned ford(float f) {
    unsigned b = __float_as_uint(f);
    return b ^ ((unsigned)((int)b >> 31) | 0x80000000u);
}

// ---------------------------------------------------------------------------
// Main fused kernel: one block per m.
//   GEMM  y[K,64] = wn2[K,c] x X[c,64]  via v_wmma_f32_16x16x4_f32,
//   branchless keyed argmax, then epilogue (sim map + integrate).
// ---------------------------------------------------------------------------
__global__ __launch_bounds__(128) void cosine_top1_kernel(
    const float* __restrict__ x,      // [m][c][64]
    const float* __restrict__ w2,     // [K][c]   original
    const float* __restrict__ wn2,    // [K][c]   row-normalized
    float* __restrict__ out_int,      // [m][256]
    float* __restrict__ out_sim,      // [m][64]
    int K) {
    __shared__ float xt[PIX * CS];    // X transposed: xt[p][c]
    __shared__ float inv_xn[PIX];
    __shared__ float wrow_n[CDIM];
    __shared__ float wrow_o[CDIM];
    __shared__ float smap[PIX];
    __shared__ unsigned long long redk[4];
    __shared__ int bcast_flat;

    const int m   = blockIdx.x;
    const int tid = threadIdx.x;      // 128 threads = 4 waves
    const float* xg = x + (size_t)m * (CDIM * PIX);

    // ---- stage X into LDS, transposed ----
    const float4* xg4 = (const float4*)xg;
    #pragma unroll
    for (int j = 0; j < (CDIM * PIX / 4) / 128; ++j) {   // 32 iters
        int i  = tid + j * 128;          // float4 index over [c][p]
        int ci = i >> 4;                 // channel
        int p0 = (i & 15) * 4;           // pixel base
        float4 v = xg4[i];
        xt[(p0 + 0) * CS + ci] = v.x;
        xt[(p0 + 1) * CS + ci] = v.y;
        xt[(p0 + 2) * CS + ci] = v.z;
        xt[(p0 + 3) * CS + ci] = v.w;
    }
    __syncthreads();

    if (tid < PIX) {
        float s = 0.f;
        for (int cc = 0; cc < CDIM; ++cc) {
            float t = xt[tid * CS + cc];
            s += t * t;
        }
        inv_xn[tid] = 1.0f / sqrtf(s);
    }
    __syncthreads();

    const int lane = tid & 31;
    const int wave = tid >> 5;
    const int ln   = lane & 15;
    const int hi   = lane >> 4;       // 0 | 1
    const int hi2  = hi * 2;          // K-pair base {0,1} vs {2,3}
    const int nKT  = (K + 15) >> 4;   // 63

    unsigned long long best = 0ull;   // (ford(sim) << 32) | ~flat

    float ixn[4];
    #pragma unroll
    for (int nt = 0; nt < 4; ++nt) ixn[nt] = inv_xn[nt * 16 + ln];

    for (int kt = wave; kt < nKT; kt += 4) {
        int rowA = kt * 16 + ln;
        if (rowA > K - 1) rowA = K - 1;                  // clamp (masked later)
        const float* ap = wn2 + (size_t)rowA * CDIM + hi2;

        v8f acc[4] = {};
        #pragma unroll 4
        for (int ch = 0; ch < CDIM / 4; ++ch) {
            v2f a = *(const v2f*)(ap + ch * 4);          // A frag: 8B global load
            int kk = ch * 4 + hi2;
            #pragma unroll
            for (int nt = 0; nt < 4; ++nt) {
                // B frag: contiguous pair from LDS (8B, conflict-free)
                v2f b = *(const v2f*)(&xt[(nt * 16 + ln) * CS + kk]);
                acc[nt] = __builtin_amdgcn_wmma_f32_16x16x4_f32(
                    false, a, false, b, (short)0, acc[nt], false, false);
            }
        }

        // ---- branchless argmax update ----
        // Lane holds rows (rbase + v), v=0..7, at pixel nt*16+ln.
        // ixn > 0 is constant within a group -> tournament on raw acc values.
        const int rbase = kt * 16 + hi * 8;
        #pragma unroll
        for (int nt = 0; nt < 4; ++nt) {
            float bval = acc[nt][0];
            int   bvi  = 0;
            #pragma unroll
            for (int v = 1; v < 8; ++v) {          // strict > keeps lowest v on ties
                bool g = acc[nt][v] > bval;
                bval = g ? acc[nt][v] : bval;
                bvi  = g ? v : bvi;
            }
            int   row = rbase + bvi;
            float sim = bval * ixn[nt];
            unsigned nflat = ~(unsigned)(row * PIX + (nt * 16 + ln));
            unsigned long long key =
                ((unsigned long long)ford(sim) << 32) | (unsigned long long)nflat;
            key  = (row < K) ? key : 0ull;         // kill clamped OOB duplicates
            best = (key > best) ? key : best;
        }
    }

    // ---- wave-level keyed-argmax reduction (wave32 butterfly, u64 max) ----
    #pragma unroll
    for (int off = 16; off; off >>= 1) {
        unsigned lo = (unsigned)best, hi32 = (unsigned)(best >> 32);
        unsigned olo = __shfl_xor(lo, off, 32);
        unsigned ohi = __shfl_xor(hi32, off, 32);
        unsigned long long o = ((unsigned long long)ohi << 32) | olo;
        best = (o > best) ? o : best;
    }
    if (lane == 0) redk[wave] = best;
    __syncthreads();
    if (tid == 0) {
        unsigned long long b = redk[0];
        #pragma unroll
        for (int i = 1; i < 4; ++i) b = (redk[i] > b) ? redk[i] : b;
        bcast_flat = (int)(~(unsigned)b);          // decode flat index
    }
    __syncthreads();

    const int flat  = bcast_flat;
    const int kstar = flat >> 6;      // / 64
    const int pstar = flat & 63;      // % 64

    // ---- epilogue: stage winning rows ----
    wrow_n[tid]       = wn2[(size_t)kstar * CDIM + tid];
    wrow_n[tid + 128] = wn2[(size_t)kstar * CDIM + tid + 128];
    wrow_o[tid]       = w2[(size_t)kstar * CDIM + tid];
    wrow_o[tid + 128] = w2[(size_t)kstar * CDIM + tid + 128];
    __syncthreads();

    // similarity map of winning codebook row (normalized)
    if (tid < PIX) {
        float s = 0.f;
        for (int cc = 0; cc < CDIM; ++cc) s += wrow_n[cc] * xt[tid * CS + cc];
        float sm = s * inv_xn[tid];
        smap[tid] = sm;
        out_sim[(size_t)m * PIX + tid] = sm;
    }
    __syncthreads();

    // integrate = x[:, p*] * s  +  w2[k*] * (1 - s)
    const float ms = smap[pstar];
    {
        int c0 = tid, c1 = tid + 128;
        out_int[(size_t)m * CDIM + c0] =
            xt[pstar * CS + c0] * ms + wrow_o[c0] * (1.0f - ms);
        out_int[(size_t)m * CDIM + c1] =
            xt[pstar * CS + c1] * ms + wrow_o[c1] * (1.0f - ms);
    }
}

// ---------------------------------------------------------------------------
extern "C" void kernel_launch(void* const* d_in, const int* in_sizes, int n_in,
                              void* d_out, int out_size, void* d_ws, size_t ws_size,
                              hipStream_t stream) {
    const float* x = (const float*)d_in[0];   // [n,L,c,h,w] fp32
    const float* w = (const float*)d_in[1];   // [K,c,1,1]   fp32
    const int K = in_sizes[1] / CDIM;               // 1000
    const int m = in_sizes[0] / (CDIM * PIX);       // 2048

    float* wn2 = (float*)d_ws;                      // K*256 floats scratch
    float* out_int = (float*)d_out;                 // [m,256]
    float* out_sim = out_int + (size_t)m * CDIM;    // [m,64]

    wnorm_prep_kernel<<<K, CDIM, 0, stream>>>(w, wn2);
    cosine_top1_kernel<<<m, 128, 0, stream>>>(x, w, wn2, out_int, out_sim, K);
}